// gnnet_19533511262572
// MI455X (gfx1250) — compile-verified
//
#include <hip/hip_runtime.h>

// ---------------------------------------------------------------------------
// GINE GNN forward for MI455X (gfx1250, wave32).
// fp32 WMMA (V_WMMA_F32_16X16X4_F32) for all dense GEMMs (exact fp32 math,
// matching the reference), float global atomics for scatter-add aggregation.
// GEMM has a mask-free fast path for fully in-bounds 64x16 tiles.
// ---------------------------------------------------------------------------

typedef __attribute__((ext_vector_type(2))) float v2f;
typedef __attribute__((ext_vector_type(8))) float v8f;

#define D_     200
#define TWO_D  400
#define L_     10
#define FEAT_  200
#define EPS_   1e-5f
#define MT     4          // 4 row-subtiles of 16 per wave => 64x16 C tile

__device__ __forceinline__ void atomAddF(float* p, float v) {
  __hip_atomic_fetch_add(p, v, __ATOMIC_RELAXED, __HIP_MEMORY_SCOPE_AGENT);
}

__global__ void k_zero(float* __restrict__ p, int n) {
  int i = blockIdx.x * blockDim.x + threadIdx.x;
  if (i < n) p[i] = 0.f;
}

// h[n,d] = x_emb1[x[n,0],d] + x_emb2[x[n,1],d]
__global__ void k_init_h(const int* __restrict__ x, const float* __restrict__ e1,
                         const float* __restrict__ e2, float* __restrict__ h, int N) {
  int i = blockIdx.x * blockDim.x + threadIdx.x;
  if (i >= N * D_) return;
  int n = i / D_, d = i % D_;
  h[i] = e1[x[n * 2 + 0] * D_ + d] + e2[x[n * 2 + 1] * D_ + d];
}

// aggr = h + self_emb   (self loop attr = (4,0))
__global__ void k_aggr_init(const float* __restrict__ h, const float* __restrict__ ee1l,
                            const float* __restrict__ ee2l, float* __restrict__ aggr, int N) {
  int i = blockIdx.x * blockDim.x + threadIdx.x;
  if (i >= N * D_) return;
  int d = i % D_;
  aggr[i] = h[i] + ee1l[4 * D_ + d] + ee2l[d];
}

// aggr[dst] += h[src] + edge_emb1[attr0] + edge_emb2[attr1]
__global__ void k_scatter(const float* __restrict__ h, const int* __restrict__ ei,
                          const int* __restrict__ ea, const float* __restrict__ ee1l,
                          const float* __restrict__ ee2l, float* __restrict__ aggr, int E) {
  int i = blockIdx.x * blockDim.x + threadIdx.x;
  if (i >= E * D_) return;
  int e = i / D_, d = i % D_;
  int s  = ei[e];
  int t  = ei[E + e];
  int a0 = ea[e * 2 + 0];
  int a1 = ea[e * 2 + 1];
  float v = h[(size_t)s * D_ + d] + ee1l[a0 * D_ + d] + ee2l[a1 * D_ + d];
  atomAddF(&aggr[(size_t)t * D_ + d], v);
}

// C[M,Nout] = act(A[M,K] @ B[K,Nout] + bias), fp32 WMMA 16x16x4.
// One wave computes a 64x16 tile (MT=4 wmma per k-step, B operand reused).
// Wave-uniform fast path for fully in-bounds tiles (no masking VALU in the
// hot loop); masked fallback for edge tiles keeps EXEC all-1s via selects.
__global__ void __launch_bounds__(256)
k_gemm(const float* __restrict__ A, const float* __restrict__ B,
       const float* __restrict__ bias, float* __restrict__ C,
       int M, int K, int Nout, int mtiles, int ntiles, int act) {
  int wave = (int)((blockIdx.x * blockDim.x + threadIdx.x) >> 5);
  int lane = threadIdx.x & 31;
  if (wave >= mtiles * ntiles) return;      // uniform per wave -> full EXEC below
  int mt = wave % mtiles;
  int nt = wave / mtiles;
  int rowbase = mt * (16 * MT);
  int colbase = nt * 16;
  int half = lane >> 4;                     // selects K pair (0/1 vs 2/3)
  int l16  = lane & 15;
  int col  = colbase + l16;

  v8f acc[MT] = {};

  bool fullTile = (rowbase + 16 * MT <= M) && (colbase + 16 <= Nout);

  if (fullTile) {
    // -------- fast path: pointer-bumped, mask-free --------
    const float* ap[MT];
#pragma unroll
    for (int m = 0; m < MT; ++m)
      ap[m] = A + (size_t)(rowbase + m * 16 + l16) * K + 2 * half;
    const float* bp = B + (size_t)(2 * half) * Nout + col;
    const size_t bstep = (size_t)4 * Nout;

    for (int kb = 0; kb < K; kb += 4) {
      v2f b;
      b.x = bp[0];
      b.y = bp[Nout];
#pragma unroll
      for (int m = 0; m < MT; ++m) {
        v2f a = *(const v2f*)ap[m];         // 8B load, k-contiguous
        __builtin_prefetch(ap[m] + 16, 0, 3);
        acc[m] = __builtin_amdgcn_wmma_f32_16x16x4_f32(
            false, a, false, b, (short)0, acc[m], false, false);
        ap[m] += 4;
      }
      bp += bstep;
    }

    float bv = bias[col];
#pragma unroll
    for (int m = 0; m < MT; ++m) {
#pragma unroll
      for (int i = 0; i < 8; ++i) {
        int row = rowbase + m * 16 + i + 8 * half;   // C layout: vgpr i, lane half
        float v = acc[m][i] + bv;
        if (act) v = v > 0.f ? v : 0.f;
        C[(size_t)row * Nout + col] = v;
      }
    }
  } else {
    // -------- edge path: branch-free per-lane masking --------
    int colc = col < Nout ? col : Nout - 1;
    float colscale = col < Nout ? 1.f : 0.f;

    for (int kb = 0; kb < K; kb += 4) {
      int kk = kb + 2 * half;
      v2f b;
      b.x = B[(size_t)kk * Nout + colc] * colscale;
      b.y = B[(size_t)(kk + 1) * Nout + colc] * colscale;
#pragma unroll
      for (int m = 0; m < MT; ++m) {
        int row  = rowbase + m * 16 + l16;
        int rowc = row < M ? row : M - 1;
        v2f a = *(const v2f*)(A + (size_t)rowc * K + kk);
        float rs = row < M ? 1.f : 0.f;
        a.x *= rs; a.y *= rs;
        acc[m] = __builtin_amdgcn_wmma_f32_16x16x4_f32(
            false, a, false, b, (short)0, acc[m], false, false);
      }
    }

    if (col < Nout) {
      float bv = bias[col];
#pragma unroll
      for (int m = 0; m < MT; ++m) {
#pragma unroll
        for (int i = 0; i < 8; ++i) {
          int row = rowbase + m * 16 + i + 8 * half;
          if (row < M) {
            float v = acc[m][i] + bv;
            if (act) v = v > 0.f ? v : 0.f;
            C[(size_t)row * Nout + col] = v;
          }
        }
      }
    }
  }
}

// partial per-channel sum / sumsq over a row chunk, atomics into stats[2*D]
__global__ void k_bn_partial(const float* __restrict__ h, float* __restrict__ stats,
                             int N, int rpb) {
  int d = threadIdx.x;
  if (d >= D_) return;
  int r0 = blockIdx.x * rpb;
  int r1 = r0 + rpb; if (r1 > N) r1 = N;
  float s = 0.f, sq = 0.f;
  for (int r = r0; r < r1; ++r) {
    float v = h[(size_t)r * D_ + d];
    s += v; sq += v * v;
  }
  atomAddF(&stats[d], s);
  atomAddF(&stats[D_ + d], sq);
}

__global__ void k_bn_apply(float* __restrict__ h, const float* __restrict__ stats,
                           const float* __restrict__ gamma, const float* __restrict__ beta,
                           int N, int doRelu) {
  int i = blockIdx.x * blockDim.x + threadIdx.x;
  if (i >= N * D_) return;
  int d = i % D_;
  float invN = 1.f / (float)N;
  float mu  = stats[d] * invN;
  float var = stats[D_ + d] * invN - mu * mu;
  float v = (h[i] - mu) * rsqrtf(var + EPS_) * gamma[d] + beta[d];
  if (doRelu) v = fmaxf(v, 0.f);
  h[i] = v;
}

__global__ void k_pool_scatter(const float* __restrict__ h, const int* __restrict__ batch,
                               float* __restrict__ pool, float* __restrict__ counts, int N) {
  int i = blockIdx.x * blockDim.x + threadIdx.x;
  if (i >= N * D_) return;
  int n = i / D_, d = i % D_;
  int g = batch[n];
  atomAddF(&pool[(size_t)g * D_ + d], h[i]);
  if (d == 0) atomAddF(&counts[g], 1.f);
}

__global__ void k_pool_div(float* __restrict__ pool, const float* __restrict__ counts, int Gn) {
  int i = blockIdx.x * blockDim.x + threadIdx.x;
  if (i >= Gn * D_) return;
  int g = i / D_;
  float c = counts[g];
  c = c > 1.f ? c : 1.f;
  pool[i] = pool[i] / c;
}

extern "C" void kernel_launch(void* const* d_in, const int* in_sizes, int n_in,
                              void* d_out, int out_size, void* d_ws, size_t ws_size,
                              hipStream_t stream) {
  const int*   x     = (const int*)d_in[0];
  const int*   ei    = (const int*)d_in[1];
  const int*   ea    = (const int*)d_in[2];
  const int*   batch = (const int*)d_in[3];
  const float* xe1   = (const float*)d_in[4];
  const float* xe2   = (const float*)d_in[5];
  const float* ee1   = (const float*)d_in[6];   // [L,5,D]
  const float* ee2   = (const float*)d_in[7];   // [L,3,D]
  const float* w1    = (const float*)d_in[8];   // [L,D,2D]
  const float* b1    = (const float*)d_in[9];   // [L,2D]
  const float* w2    = (const float*)d_in[10];  // [L,2D,D]
  const float* b2    = (const float*)d_in[11];  // [L,D]
  const float* gam   = (const float*)d_in[12];  // [L,D]
  const float* bet   = (const float*)d_in[13];  // [L,D]
  const float* fw    = (const float*)d_in[14];  // [D,FEAT]
  const float* fb    = (const float*)d_in[15];
  const float* ow1   = (const float*)d_in[16];  // [FEAT,FEAT]
  const float* ob1   = (const float*)d_in[17];
  const float* ow2   = (const float*)d_in[18];  // [FEAT,FEAT/2]
  const float* ob2   = (const float*)d_in[19];
  float* out = (float*)d_out;

  const int N = in_sizes[0] / 2;
  const int E = in_sizes[1] / 2;
  const int G = 2500;

  // workspace layout (floats)
  float* ws     = (float*)d_ws;
  float* h      = ws;                              // N*D
  float* aggr   = h    + (size_t)N * D_;           // N*D
  float* h1     = aggr + (size_t)N * D_;           // N*2D
  float* stats  = h1   + (size_t)N * TWO_D;        // 2*D
  float* pool   = stats + 2 * D_;                  // G*D
  float* counts = pool + (size_t)G * D_;           // G
  float* t1     = counts + G;                      // G*FEAT

  const int TB = 256;
  int gridND = (N * D_ + TB - 1) / TB;
  int gridED = (E * D_ + TB - 1) / TB;

  auto launch_gemm = [&](const float* A, const float* B, const float* bias, float* C,
                         int M, int K, int Nout, int act) {
    int mtiles = (M + 16 * MT - 1) / (16 * MT);
    int ntiles = (Nout + 15) / 16;
    int waves  = mtiles * ntiles;
    int blocks = (waves + 7) / 8;                  // 8 waves / 256-thread block
    k_gemm<<<blocks, TB, 0, stream>>>(A, B, bias, C, M, K, Nout, mtiles, ntiles, act);
  };

  k_init_h<<<gridND, TB, 0, stream>>>(x, xe1, xe2, h, N);

  for (int l = 0; l < L_; ++l) {
    const float* ee1l = ee1 + (size_t)l * 5 * D_;
    const float* ee2l = ee2 + (size_t)l * 3 * D_;

    k_aggr_init<<<gridND, TB, 0, stream>>>(h, ee1l, ee2l, aggr, N);
    k_scatter<<<gridED, TB, 0, stream>>>(h, ei, ea, ee1l, ee2l, aggr, E);

    launch_gemm(aggr, w1 + (size_t)l * D_ * TWO_D, b1 + (size_t)l * TWO_D,
                h1, N, D_, TWO_D, /*relu=*/1);
    launch_gemm(h1, w2 + (size_t)l * TWO_D * D_, b2 + (size_t)l * D_,
                h, N, TWO_D, D_, /*relu=*/0);

    k_zero<<<(2 * D_ + TB - 1) / TB, TB, 0, stream>>>(stats, 2 * D_);
    int rpb = 256;
    k_bn_partial<<<(N + rpb - 1) / rpb, TB, 0, stream>>>(h, stats, N, rpb);
    k_bn_apply<<<gridND, TB, 0, stream>>>(h, stats, gam + (size_t)l * D_,
                                          bet + (size_t)l * D_, N, l < L_ - 1);
  }

  // global mean pool
  k_zero<<<(G * D_ + TB - 1) / TB, TB, 0, stream>>>(pool, G * D_);
  k_zero<<<(G + TB - 1) / TB, TB, 0, stream>>>(counts, G);
  k_pool_scatter<<<gridND, TB, 0, stream>>>(h, batch, pool, counts, N);
  k_pool_div<<<(G * D_ + TB - 1) / TB, TB, 0, stream>>>(pool, counts, G);

  // heads: h_global = pool @ feat_w + fb  -> out[0 : G*FEAT]
  launch_gemm(pool, fw, fb, out, G, D_, FEAT_, 0);
  // t1 = relu(h_global @ out_w1 + ob1)
  launch_gemm(out, ow1, ob1, t1, G, FEAT_, FEAT_, 1);
  // out tuple tail: same tensor twice
  launch_gemm(t1, ow2, ob2, out + (size_t)G * FEAT_, G, FEAT_, FEAT_ / 2, 0);
  launch_gemm(t1, ow2, ob2, out + (size_t)G * FEAT_ + (size_t)G * (FEAT_ / 2),
              G, FEAT_, FEAT_ / 2, 0);
}